// FlowLenia_63282048139474
// MI455X (gfx1250) — compile-verified
//
#include <hip/hip_runtime.h>
#include <hip/hip_bf16.h>
#include <math.h>

// ---------------------------------------------------------------------------
// FlowLenia on MI455X (gfx1250), compile-only tuning round 3.
//  - DFT-as-GEMM on the fp32 WMMA pipe (v_wmma_f32_16x16x4_f32, wave32)
//  - A-operand slab (16x256 fp32) staged into LDS via the Tensor Data Mover
//    (tensor_load_to_lds + s_wait_tensorcnt), 8x reuse across the block
//  - templated DUAL (no runtime branch in the K loop) and BSYM (symmetric W
//    operand read row-wise -> contiguous global_load_b64)
//  - B fragments software-pipelined one K-step ahead
//  - reintegration halo staged to LDS via global_load_async_to_lds_b32
// ---------------------------------------------------------------------------

typedef float v2f __attribute__((ext_vector_type(2)));
typedef float v8f __attribute__((ext_vector_type(8)));
typedef unsigned int u32x4 __attribute__((ext_vector_type(4)));
typedef int i32x4 __attribute__((ext_vector_type(4)));
typedef int i32x8 __attribute__((ext_vector_type(8)));

#define PL 65536          // one 256x256 fp32 plane (in floats)

// plane offsets inside workspace (units of PL floats)
#define WRp   0
#define WIp   1
#define Pp    2    // 13 planes: A(0..2), fftshift(nK)(3..12)
#define TRp   15   // 13
#define TIp   28   // 13
#define FRp   41   // 13: fA real (0..2), fK real (3..12)
#define FIp   54   // 13
#define GRp   67   // 10
#define GIp   77   // 10
#define HRp   87   // 10
#define HIp   97   // 10
#define URp   107  // 10: real(ifft2)
#define UCp   117  // 3 : per-color growth sums
#define MUp   120  // 6 : mu planes, [d*3+c], d=0 -> y, d=1 -> x
#define ASUMp 126  // 1 : sum_c A

#define SIG   0.65f
#define DTc   0.2f
#define MAc   (5.0f - SIG)
#define INV4S2 (1.0f / (4.0f * SIG * SIG))

// ----- feature probes -------------------------------------------------------
#if __has_builtin(__builtin_amdgcn_tensor_load_to_lds) && \
    __has_builtin(__builtin_amdgcn_s_wait_tensorcnt)
#define HAVE_TDM 1
#else
#define HAVE_TDM 0
#endif

#if __has_builtin(__builtin_amdgcn_global_load_async_to_lds_b32) && \
    __has_builtin(__builtin_amdgcn_s_wait_asynccnt)
#define HAVE_ASYNC 1
#else
#define HAVE_ASYNC 0
#endif

#if HAVE_TDM
// One 2-D TDM descriptor: load a 16-row x 256-col fp32 slab (row stride 256
// floats) from global into LDS. Block-uniform; issue from one wave only.
__device__ __forceinline__ void tdm_load_slab16x256(const float* gsrc,
                                                    float* lds_dst) {
  unsigned long long ga = (unsigned long long)(uintptr_t)gsrc;
  unsigned int la = (unsigned int)(uintptr_t)(void*)lds_dst;  // LDS byte offset
  u32x4 g0;
  g0[0] = 1u;                                           // count=1 (valid D#)
  g0[1] = la;                                           // lds_addr
  g0[2] = (unsigned int)(ga & 0xffffffffu);             // global_addr[31:0]
  g0[3] = (unsigned int)((ga >> 32) & 0x1ffffffu)       // global_addr[56:32]
          | (2u << 30);                                 // type=2 (image)
  i32x8 g1;
  g1[0] = (int)(2u << 16);          // data_size=2 (4 bytes), wg_mask=0
  g1[1] = (int)(256u << 16);        // tensor_dim0[15:0]=256 (atomic_addr=0)
  g1[2] = 0;                        // tensor_dim0 hi / tensor_dim1 lo (65536)
  g1[3] = (int)(1u | (256u << 16)); // tensor_dim1 hi=1 ; tile_dim0=256
  g1[4] = 16;                       // tile_dim1=16 ; tile_dim2=0
  g1[5] = 256;                      // tensor_dim0_stride lo32 = 256
  g1[6] = 0;                        // stride0 hi / stride1 lo
  g1[7] = 0;                        // stride1 hi
  i32x4 gz = {0, 0, 0, 0};
#if defined(__clang_major__) && (__clang_major__ >= 23)
  i32x8 gz8 = {0, 0, 0, 0, 0, 0, 0, 0};
  __builtin_amdgcn_tensor_load_to_lds(g0, g1, gz, gz, gz8, 0);
#else
  __builtin_amdgcn_tensor_load_to_lds(g0, g1, gz, gz, 0);
#endif
}
#endif

// ---------------------------------------------------------------------------
// 1. DFT tables: W[j][k] = exp(-2*pi*i*j*k/256), angle reduced mod 256.
__global__ void build_dft(float* __restrict__ ws) {
  int idx = blockIdx.x * blockDim.x + threadIdx.x;
  int j = idx >> 8, k = idx & 255;
  int jk = (j * k) & 255;
  float ang = -6.283185307179586f * (float)jk * (1.0f / 256.0f);
  ws[WRp * PL + idx] = __cosf(ang);
  ws[WIp * PL + idx] = __sinf(ang);
}

// ---------------------------------------------------------------------------
// 2. interleaved -> planar; fftshift(nK); Asum plane.
__global__ void pack_planar(const float* __restrict__ A,
                            const float* __restrict__ nK,
                            float* __restrict__ ws) {
  int idx = blockIdx.x * blockDim.x + threadIdx.x;
  int i = idx >> 8, j = idx & 255;
  float a0 = A[idx * 3 + 0], a1 = A[idx * 3 + 1], a2 = A[idx * 3 + 2];
  ws[(Pp + 0) * PL + idx] = a0;
  ws[(Pp + 1) * PL + idx] = a1;
  ws[(Pp + 2) * PL + idx] = a2;
  ws[ASUMp * PL + idx] = a0 + a1 + a2;
  int sidx = (((i + 128) & 255) << 8) | ((j + 128) & 255);   // fftshift
#pragma unroll
  for (int k = 0; k < 10; ++k)
    ws[(Pp + 3 + k) * PL + idx] = nK[sidx * 10 + k];
}

// ---------------------------------------------------------------------------
// 3. WMMA GEMM: Out = (A1*B1 + s2 * A2*B2) * scale, batch = blockIdx.z.
//    Block = 8 waves; blockIdx.y = tm (shared A slab), wave -> tn tile.
//    A slab (16x256, x2 if DUAL) staged in LDS via TDM (or coop fallback).
//    BSYM: B is symmetric (W), so fragments are read row-wise (contiguous).
template <bool DUAL, bool BSYM>
__global__ __launch_bounds__(256)
void gemm_wmma(const float* __restrict__ A1, int sA1,
               const float* __restrict__ B1, int sB1,
               const float* __restrict__ A2, int sA2,
               const float* __restrict__ B2, int sB2,
               float s2, float scale,
               float* __restrict__ Out, int sOut) {
  __shared__ float sAbuf[(DUAL ? 2 : 1) * 16 * 256];
  float* sSlab1 = sAbuf;
  float* sSlab2 = DUAL ? sAbuf + 16 * 256 : nullptr;

  int z = blockIdx.z;
  A1 += (size_t)z * sA1;
  B1 += (size_t)z * sB1;
  Out += (size_t)z * sOut;
  if (DUAL) { A2 += (size_t)z * sA2; B2 += (size_t)z * sB2; }

  int lane = threadIdx.x;                  // 0..31
  int wave = threadIdx.y;                  // 0..7
  int tm = blockIdx.y;                     // 0..15
  int tn = blockIdx.x * 8 + wave;          // 0..15
  int r = lane & 15;
  int kh = lane >> 4;

  const float* a1slab = A1 + (size_t)tm * 16 * 256;
  const float* a2slab = DUAL ? A2 + (size_t)tm * 16 * 256 : nullptr;

#if HAVE_TDM
  if (wave == 0) {
    tdm_load_slab16x256(a1slab, sSlab1);
    if (DUAL) tdm_load_slab16x256(a2slab, sSlab2);
    __builtin_amdgcn_s_wait_tensorcnt(0);
  }
#else
  for (int t = wave * 32 + lane; t < 16 * 256; t += 256) {
    sSlab1[t] = a1slab[t];
    if (DUAL) sSlab2[t] = a2slab[t];
  }
#endif
  __syncthreads();

  const float* aL1 = sSlab1 + r * 256 + 2 * kh;
  const float* aL2 = DUAL ? sSlab2 + r * 256 + 2 * kh : nullptr;
  const float* b1p = BSYM ? B1 + (size_t)(tn * 16 + r) * 256 + 2 * kh
                          : B1 + (size_t)(2 * kh) * 256 + tn * 16 + r;
  const float* b2p = nullptr;
  if (DUAL)
    b2p = BSYM ? B2 + (size_t)(tn * 16 + r) * 256 + 2 * kh
               : B2 + (size_t)(2 * kh) * 256 + tn * 16 + r;

  auto loadB = [&](const float* bp, int kk) {
    v2f b;
    if (BSYM) {
      b.x = bp[kk];
      b.y = bp[kk + 1];
    } else {
      b.x = bp[(size_t)kk * 256];
      b.y = bp[(size_t)kk * 256 + 256];
    }
    return b;
  };

  v8f acc = {};
  v8f acc2 = {};
  v2f b1 = loadB(b1p, 0);
  v2f b2 = {};
  if (DUAL) b2 = loadB(b2p, 0);
  for (int kk = 0; kk < 256; kk += 4) {
    // prefetch next K fragments (harmless overrun stays inside workspace)
    v2f b1n = loadB(b1p, kk + 4);
    v2f b2n = {};
    if (DUAL) b2n = loadB(b2p, kk + 4);
    v2f a1;
    a1.x = aL1[kk];
    a1.y = aL1[kk + 1];
    acc = __builtin_amdgcn_wmma_f32_16x16x4_f32(
        false, a1, false, b1, (short)0, acc, false, false);
    if (DUAL) {
      v2f a2;
      a2.x = aL2[kk];
      a2.y = aL2[kk + 1];
      acc2 = __builtin_amdgcn_wmma_f32_16x16x4_f32(
          false, a2, false, b2, (short)0, acc2, false, false);
    }
    b1 = b1n;
    b2 = b2n;
  }

  float* orow = Out + (size_t)(tm * 16 + 8 * kh) * 256 + tn * 16 + r;
#pragma unroll
  for (int v = 0; v < 8; ++v) {
    float val = acc[v];
    if (DUAL) val += s2 * acc2[v];
    orow[(size_t)v * 256] = val * scale;
  }
}

// ---------------------------------------------------------------------------
// 4. Spectrum product: G_k = fK_k * fA_{k%3}.
__global__ void spec_mul(float* __restrict__ ws) {
  int idx = blockIdx.x * blockDim.x + threadIdx.x;
  int k = blockIdx.z;
  int c = k % 3;
  float kr = ws[(FRp + 3 + k) * PL + idx], ki = ws[(FIp + 3 + k) * PL + idx];
  float ar = ws[(FRp + c) * PL + idx],     ai = ws[(FIp + c) * PL + idx];
  ws[(GRp + k) * PL + idx] = kr * ar - ki * ai;
  ws[(GIp + k) * PL + idx] = kr * ai + ki * ar;
}

// ---------------------------------------------------------------------------
// 5. Growth fn + per-color reduce.
__global__ void growth_reduce(const float* __restrict__ m,
                              const float* __restrict__ s,
                              const float* __restrict__ h,
                              float* __restrict__ ws) {
  int idx = blockIdx.x * blockDim.x + threadIdx.x;
  float acc[3] = {0.f, 0.f, 0.f};
#pragma unroll
  for (int k = 0; k < 10; ++k) {
    float u = ws[(URp + k) * PL + idx];
    float t = (u - m[k]) / s[k];
    float g = (__expf(-0.5f * t * t) * 2.0f - 1.0f) * h[k];
    acc[k % 3] += g;
  }
#pragma unroll
  for (int c = 0; c < 3; ++c) ws[(UCp + c) * PL + idx] = acc[c];
}

// ---------------------------------------------------------------------------
// 6. Sobel (zero-padded convolve2d), alpha blend, flow clip -> mu planes.
__device__ __forceinline__ float ld0(const float* p, int i, int j) {
  if ((unsigned)i > 255u || (unsigned)j > 255u) return 0.f;
  return p[(i << 8) | j];
}
__global__ void force_mu(const float* __restrict__ A, float* __restrict__ ws) {
  int idx = blockIdx.x * blockDim.x + threadIdx.x;
  int i = idx >> 8, j = idx & 255;
  const float* asum = ws + (size_t)ASUMp * PL;
  float pp = ld0(asum, i + 1, j + 1), p0 = ld0(asum, i + 1, j), pm = ld0(asum, i + 1, j - 1);
  float mp = ld0(asum, i - 1, j + 1), m0 = ld0(asum, i - 1, j), mm = ld0(asum, i - 1, j - 1);
  float zp = ld0(asum, i, j + 1), zm = ld0(asum, i, j - 1);
  float syA = pp + 2.f * p0 + pm - mp - 2.f * m0 - mm;
  float sxA = pp + 2.f * zp + mp - pm - 2.f * zm - mm;

  float py = (float)i + 0.5f, px = (float)j + 0.5f;
#pragma unroll
  for (int c = 0; c < 3; ++c) {
    const float* U = ws + (size_t)(UCp + c) * PL;
    float upp = ld0(U, i + 1, j + 1), up0 = ld0(U, i + 1, j), upm = ld0(U, i + 1, j - 1);
    float ump = ld0(U, i - 1, j + 1), um0 = ld0(U, i - 1, j), umm = ld0(U, i - 1, j - 1);
    float uzp = ld0(U, i, j + 1), uzm = ld0(U, i, j - 1);
    float syU = upp + 2.f * up0 + upm - ump - 2.f * um0 - umm;
    float sxU = upp + 2.f * uzp + ump - upm - 2.f * uzm - umm;

    float a = A[idx * 3 + c] * (1.0f / 3.0f);
    float alpha = fminf(a * a, 1.0f);
    float Fy = syU * (1.f - alpha) - syA * alpha;
    float Fx = sxU * (1.f - alpha) - sxA * alpha;
    float muy = py + fminf(fmaxf(DTc * Fy, -MAc), MAc);
    float mux = px + fminf(fmaxf(DTc * Fx, -MAc), MAc);
    muy = fminf(fmaxf(muy, SIG), 256.0f - SIG);
    mux = fminf(fmaxf(mux, SIG), 256.0f - SIG);
    ws[(MUp + 0 * 3 + c) * PL + idx] = muy;
    ws[(MUp + 1 * 3 + c) * PL + idx] = mux;
  }
}

// ---------------------------------------------------------------------------
// 7. Reintegration gather: 26x26 halo of A (3ch) + mu (6 planes) in LDS.
#if HAVE_ASYNC
__device__ __forceinline__ void async_b32(const float* g, float* l) {
  // builtin signature (from hipcc diagnostic): param0 = AS1 int*, then AS3
  // int*, then imm offset, imm cpol.
  __builtin_amdgcn_global_load_async_to_lds_b32(
      (__attribute__((address_space(1))) int*)(uintptr_t)g,
      (__attribute__((address_space(3))) int*)(uintptr_t)l, 0, 0);
}
#endif

__global__ __launch_bounds__(256)
void reint(const float* __restrict__ A, const float* __restrict__ ws,
           float* __restrict__ out) {
  __shared__ float sA[26 * 26 * 3];
  __shared__ float sMu[6][26 * 26];
  int tj = threadIdx.x, ti = threadIdx.y;        // 16x16
  int bi = blockIdx.y * 16, bj = blockIdx.x * 16;
  int t = ti * 16 + tj;
  for (int p = t; p < 26 * 26; p += 256) {
    int li = p / 26, lj = p % 26;
    int gi = (bi + li - 5) & 255, gj = (bj + lj - 5) & 255;
    int g = (gi << 8) | gj;
#if HAVE_ASYNC
    async_b32(&A[g * 3 + 0], &sA[p * 3 + 0]);
    async_b32(&A[g * 3 + 1], &sA[p * 3 + 1]);
    async_b32(&A[g * 3 + 2], &sA[p * 3 + 2]);
#pragma unroll
    for (int d = 0; d < 6; ++d) async_b32(&ws[(MUp + d) * PL + g], &sMu[d][p]);
#else
    sA[p * 3 + 0] = A[g * 3 + 0];
    sA[p * 3 + 1] = A[g * 3 + 1];
    sA[p * 3 + 2] = A[g * 3 + 2];
#pragma unroll
    for (int d = 0; d < 6; ++d) sMu[d][p] = ws[(MUp + d) * PL + g];
#endif
  }
#if HAVE_ASYNC
  __builtin_amdgcn_s_wait_asynccnt(0);
#endif
  __syncthreads();

  int i = bi + ti, j = bj + tj;
  float py = (float)i + 0.5f, px = (float)j + 0.5f;
  float acc0 = 0.f, acc1 = 0.f, acc2 = 0.f;
  for (int si = 0; si < 11; ++si) {
    int lrow = (ti + si) * 26 + tj;
    for (int sj = 0; sj < 11; ++sj) {
      int p = lrow + sj;
#pragma unroll
      for (int c = 0; c < 3; ++c) {
        float zy = 0.5f + SIG - fabsf(py - sMu[c][p]);
        float zx = 0.5f + SIG - fabsf(px - sMu[3 + c][p]);
        zy = fminf(fmaxf(zy, 0.f), 1.f);
        zx = fminf(fmaxf(zx, 0.f), 1.f);
        float area = zy * zx * INV4S2;
        float v = sA[p * 3 + c] * area;
        if (c == 0) acc0 += v; else if (c == 1) acc1 += v; else acc2 += v;
      }
    }
  }
  int g = (i << 8) | j;
  out[g * 3 + 0] = acc0;
  out[g * 3 + 1] = acc1;
  out[g * 3 + 2] = acc2;
}

// ---------------------------------------------------------------------------
extern "C" void kernel_launch(void* const* d_in, const int* in_sizes, int n_in,
                              void* d_out, int out_size, void* d_ws, size_t ws_size,
                              hipStream_t stream) {
  const float* A  = (const float*)d_in[0];
  const float* nK = (const float*)d_in[1];
  const float* m  = (const float*)d_in[2];
  const float* s  = (const float*)d_in[3];
  const float* h  = (const float*)d_in[4];
  float* out = (float*)d_out;
  float* ws  = (float*)d_ws;
  auto pl = [&](int p) { return ws + (size_t)p * PL; };

  dim3 pw_grid(256), pw_blk(256);
  build_dft<<<pw_grid, pw_blk, 0, stream>>>(ws);
  pack_planar<<<pw_grid, pw_blk, 0, stream>>>(A, nK, ws);

  dim3 gblk(32, 8);
  // Forward FFT of 13 planes: T = W * P   (A = W rows, B = data, not symmetric)
  gemm_wmma<false, false><<<dim3(2, 16, 13), gblk, 0, stream>>>(
      pl(WRp), 0, pl(Pp), PL, nullptr, 0, nullptr, 0, 0.f, 1.f, pl(TRp), PL);
  gemm_wmma<false, false><<<dim3(2, 16, 13), gblk, 0, stream>>>(
      pl(WIp), 0, pl(Pp), PL, nullptr, 0, nullptr, 0, 0.f, 1.f, pl(TIp), PL);
  // F = T * W : Fr = Tr*Wr - Ti*Wi ; Fi = Tr*Wi + Ti*Wr   (B = W, symmetric)
  gemm_wmma<true, true><<<dim3(2, 16, 13), gblk, 0, stream>>>(
      pl(TRp), PL, pl(WRp), 0, pl(TIp), PL, pl(WIp), 0, -1.f, 1.f, pl(FRp), PL);
  gemm_wmma<true, true><<<dim3(2, 16, 13), gblk, 0, stream>>>(
      pl(TRp), PL, pl(WIp), 0, pl(TIp), PL, pl(WRp), 0, +1.f, 1.f, pl(FIp), PL);

  spec_mul<<<dim3(256, 1, 10), pw_blk, 0, stream>>>(ws);

  // H = conj(W) * G : Hr = Wr*Gr + Wi*Gi ; Hi = Wr*Gi - Wi*Gr
  gemm_wmma<true, false><<<dim3(2, 16, 10), gblk, 0, stream>>>(
      pl(WRp), 0, pl(GRp), PL, pl(WIp), 0, pl(GIp), PL, +1.f, 1.f, pl(HRp), PL);
  gemm_wmma<true, false><<<dim3(2, 16, 10), gblk, 0, stream>>>(
      pl(WRp), 0, pl(GIp), PL, pl(WIp), 0, pl(GRp), PL, -1.f, 1.f, pl(HIp), PL);
  // U = Re(H * conj(W)) / N^2 = (Hr*Wr + Hi*Wi) / 65536   (B = W, symmetric)
  gemm_wmma<true, true><<<dim3(2, 16, 10), gblk, 0, stream>>>(
      pl(HRp), PL, pl(WRp), 0, pl(HIp), PL, pl(WIp), 0, +1.f,
      1.0f / 65536.0f, pl(URp), PL);

  growth_reduce<<<pw_grid, pw_blk, 0, stream>>>(m, s, h, ws);
  force_mu<<<pw_grid, pw_blk, 0, stream>>>(A, ws);
  reint<<<dim3(16, 16), dim3(16, 16), 0, stream>>>(A, ws, out);

  (void)in_sizes; (void)n_in; (void)out_size; (void)ws_size;
}